// ComplementaryGraycode_21964462752277
// MI455X (gfx1250) — compile-verified
//
#include <hip/hip_runtime.h>
#include <math.h>
#include <stdint.h>

// Native clang vector type so __builtin_nontemporal_load/store apply directly.
typedef float f32x4 __attribute__((ext_vector_type(4)));

namespace {
constexpr int   kNImgs   = 24;                          // 2*PS + 2*GC planes
constexpr float kBgThr2  = 0.01f;   // (2*0.05)^2: 0.5*sqrt(ss) > 0.05 <=> ss > 0.01
constexpr float kTwoPi   = 6.28318530717958647692f;
constexpr float kHalfPi  = 1.57079632679489661923f;
constexpr float k3HalfPi = 4.71238898038468985769f;
constexpr float kScale   = 16.0f / kTwoPi;              // WAVELENGTH / (2*pi)

// Pin a uniform pointer into an SGPR pair and make it opaque to
// reassociation/CSE, so each access selects the saddr global form:
//   global_load_b128 vdst, v_off32, s[base:base+1]
// instead of per-lane 64-bit VGPR address arithmetic.
__device__ __forceinline__ const char* sbase(const char* p) {
  asm("" : "+s"(p));
  return p;
}
__device__ __forceinline__ char* sbase_mut(char* p) {
  asm("" : "+s"(p));
  return p;
}

__device__ __forceinline__ float unwrap_one(float p, int v2, float mask) {
  // v2 = 8-bit binary code (MSB-first) from the graycode scan.
  // dec_k1 = sum(gi[:-1]*ratio[1:]) == v2 >> 1 ; dec_k2 = (v2 + 1) // 2.
  const int k1 = v2 >> 1;
  const int k2 = (v2 + 1) >> 1;
  float add;
  if (p <= kHalfPi)       add = kTwoPi * (float)k2;            // m1
  else if (p >= k3HalfPi) add = kTwoPi * (float)k2 - kTwoPi;   // m3
  else                    add = kTwoPi * (float)k1;            // m2
  return (p + add) * mask * kScale;
}

// Gray -> binary for an 8-bit MSB-first code: bit(7-k) of the result is the
// XOR-prefix raw[0]^...^raw[k], i.e. exactly the reference's associative scan.
__device__ __forceinline__ int gray_fold8(int g) {
  g ^= g >> 1;
  g ^= g >> 2;
  g ^= g >> 4;
  return g;
}
} // namespace

__global__ __launch_bounds__(256)
void cgc_decode_kernel(const float* __restrict__ img, float* __restrict__ out,
                       int n4 /* = H*W/4, float4 groups per plane */) {
  const uint32_t q = blockIdx.x * blockDim.x + threadIdx.x;
  if (q >= (uint32_t)n4) return;

  const size_t   plane = (size_t)n4 * sizeof(f32x4);   // bytes per image plane
  const uint32_t voff  = q * (uint32_t)sizeof(f32x4);  // shared 32-bit voffset

  // Per-plane uniform SGPR bases (SALU-computed, co-executes with VALU).
  const char* pb[24];
  #pragma unroll
  for (int k = 0; k < 24; ++k)
    pb[k] = sbase((const char*)img + (size_t)k * plane);

  // Prefetch the 16 graycode planes (global_prefetch_b8, near scope so the
  // lines land in WGP$+L2); overlaps the atan2-heavy phase math below.
  #pragma unroll
  for (int k = 0; k < 16; ++k)
    __builtin_prefetch((const void*)(pb[8 + k] + voff), 0, 3);

  // ---- phase-shift planes: 8 coalesced b128 saddr loads, NT stream -------
  f32x4 psv[8];
  #pragma unroll
  for (int k = 0; k < 8; ++k)
    psv[k] = __builtin_nontemporal_load((const f32x4*)(pb[k] + voff));

  // delta = [0, pi/2, pi, 3pi/2] => sin = [0,1,0,-1], cos = [1,0,-1,0]
  const f32x4 s_col = psv[1] - psv[3];
  const f32x4 c_col = psv[0] - psv[2];
  const f32x4 s_row = psv[5] - psv[7];
  const f32x4 c_row = psv[4] - psv[6];
  const f32x4 thr =
      (((psv[0] + psv[1]) + (psv[2] + psv[3])) +
       ((psv[4] + psv[5]) + (psv[6] + psv[7]))) * 0.125f;   // mean of 8

  float pcol[4], prow[4], maskv[4];
  #pragma unroll
  for (int c = 0; c < 4; ++c) {
    // mask: 0.5*sqrt(s^2+c^2) > 0.05  <=>  s^2+c^2 > 0.01  (no sqrt needed)
    const float ssc = s_col[c] * s_col[c] + c_col[c] * c_col[c];
    const float ssr = s_row[c] * s_row[c] + c_row[c] * c_row[c];
    maskv[c] = (ssc > kBgThr2 || ssr > kBgThr2) ? 1.0f : 0.0f;
    const float pc = -atan2f(s_col[c], c_col[c]);
    const float pr = -atan2f(s_row[c], c_row[c]);
    pcol[c] = (pc < 0.0f) ? pc + kTwoPi : pc;
    prow[c] = (pr < 0.0f) ? pr + kTwoPi : pr;
  }

  // ---- graycode: pack raw threshold bits MSB-first, fold to binary -------
  int rc[4] = {0, 0, 0, 0}, rr[4] = {0, 0, 0, 0};
  #pragma unroll
  for (int k = 0; k < 8; ++k) {
    const f32x4 gcv = __builtin_nontemporal_load((const f32x4*)(pb[8 + k]  + voff));
    const f32x4 grv = __builtin_nontemporal_load((const f32x4*)(pb[16 + k] + voff));
    #pragma unroll
    for (int c = 0; c < 4; ++c) {
      rc[c] = (rc[c] << 1) | ((gcv[c] > thr[c]) ? 1 : 0);
      rr[c] = (rr[c] << 1) | ((grv[c] > thr[c]) ? 1 : 0);
    }
  }

  // ---- unwrap + scale, interleave to (H, W, 2), two b128 NT stores -------
  f32x4 o0, o1;
  o0.x = unwrap_one(pcol[0], gray_fold8(rc[0]), maskv[0]);
  o0.y = unwrap_one(prow[0], gray_fold8(rr[0]), maskv[0]);
  o0.z = unwrap_one(pcol[1], gray_fold8(rc[1]), maskv[1]);
  o0.w = unwrap_one(prow[1], gray_fold8(rr[1]), maskv[1]);
  o1.x = unwrap_one(pcol[2], gray_fold8(rc[2]), maskv[2]);
  o1.y = unwrap_one(prow[2], gray_fold8(rr[2]), maskv[2]);
  o1.z = unwrap_one(pcol[3], gray_fold8(rc[3]), maskv[3]);
  o1.w = unwrap_one(prow[3], gray_fold8(rr[3]), maskv[3]);

  char* ob = sbase_mut((char*)out);
  const uint32_t ooff = q * 32u;               // two float4s per thread
  __builtin_nontemporal_store(o0, (f32x4*)(ob + ooff));
  __builtin_nontemporal_store(o1, (f32x4*)(ob + ooff + 16));
}

extern "C" void kernel_launch(void* const* d_in, const int* in_sizes, int n_in,
                              void* d_out, int out_size, void* d_ws, size_t ws_size,
                              hipStream_t stream) {
  (void)n_in; (void)out_size; (void)d_ws; (void)ws_size;
  const float* img = (const float*)d_in[0];
  float* out = (float*)d_out;
  const int npix = in_sizes[0] / kNImgs;  // H*W = 2048*2448 = 5,013,504
  const int n4   = npix / 4;              // divisible by 4 for 2048x2448
  const int block = 256;                  // 8 wave32s per block
  const int grid  = (n4 + block - 1) / block;
  cgc_decode_kernel<<<grid, block, 0, stream>>>(img, out, n4);
}